// Discriminator_42219528520158
// MI455X (gfx1250) — compile-verified
//
#include <hip/hip_runtime.h>
#include <math.h>

// CDNA5 / gfx1250: wave32, WMMA 16x16x4 f32
typedef float v2f __attribute__((ext_vector_type(2)));
typedef float v8f __attribute__((ext_vector_type(8)));

#define DD     2688      // L * N_STATES
#define BATCH  64
#define NTILES 168       // DD / 16
#define KROWS_PER_WAVE 672   // DD / 4 waves
#define PF_ROWS 32           // prefetch distance in K-rows (8 K-steps)

// ---------------------------------------------------------------------------
// Kernel 0: zero the small accumulation workspace (65 floats)
// ---------------------------------------------------------------------------
__global__ void potts_init_ws(float* ws) {
    int t = blockIdx.x * blockDim.x + threadIdx.x;
    if (t < 65) ws[t] = 0.0f;
}

// One K-step (4 rows of J): 1 B-fragment + 4 M-tile WMMAs.
// PF: compile-time prefetch toggle so the hot loop stays branch-free.
template <bool PF>
__device__ __forceinline__ void kstep(const float*& jp, const float*& xp,
                                      v8f acc[4], float& jsq) {
    if (PF) __builtin_prefetch(jp + (size_t)PF_ROWS * DD, 0, 3);

    const float b0 = jp[0];
    const float b1 = jp[DD];
    v2f bfrag; bfrag[0] = b0; bfrag[1] = b1;
    jsq += b0 * b0 + b1 * b1;          // fused reg_j (each J element once)

#pragma unroll
    for (int m = 0; m < 4; ++m) {
        const float2 af =
            *reinterpret_cast<const float2*>(xp + (size_t)m * 16 * DD);
        v2f afrag; afrag[0] = af.x; afrag[1] = af.y;
        acc[m] = __builtin_amdgcn_wmma_f32_16x16x4_f32(
            false, afrag, false, bfrag, (short)0, acc[m], false, false);
    }
    jp += 4 * (size_t)DD;              // advance 4 K-rows
    xp += 4;
}

// ---------------------------------------------------------------------------
// Kernel 1: Y = Xf @ J via V_WMMA_F32_16X16X4_F32, fused with
//           per-row bilinear contraction  sum_n Y[b,n]*Xf[b,n]  and reg_j.
// Grid: 168 blocks (one 16-col N-tile of J each), 128 threads = 4 waves.
// Each wave covers a contiguous 672-row K-chunk; J is read exactly once.
// ---------------------------------------------------------------------------
__global__ __launch_bounds__(128) void potts_gemm_kernel(
    const float* __restrict__ X,     // [64, DD] row-major
    const float* __restrict__ J,     // [DD, DD] row-major
    float* __restrict__ ws_energy,   // [64] partial j_sum accumulators
    float* __restrict__ ws_regj)     // [1]
{
    __shared__ float lds_y[4][4][256];   // [wave][mtile][16x16 row-major]
    __shared__ float lds_jsq[128];

    const int ntile = blockIdx.x;
    const int col0  = ntile * 16;
    const int tid   = (int)threadIdx.x;
    const int wave  = tid >> 5;
    const int lane  = tid & 31;
    const int half  = lane >> 4;     // lane half selects K+0/1 vs K+2/3
    const int lr    = lane & 15;     // row (A) / column (B) within tile

    v8f acc[4] = {};                 // 4 M-tiles x (16x16 f32) accumulators
    float jsq = 0.0f;

    const int kBegin = wave * KROWS_PER_WAVE;

    // B walks down J's rows in this block's 16-column stripe;
    // A walks along Xf row `lr` (per-m 16*DD offsets fold into immediates).
    const float* jp = J + (size_t)(kBegin + 2 * half) * DD + col0 + lr;
    const float* xp = X + (size_t)lr * DD + kBegin + 2 * half;

    // Main loop: unconditional near-scope prefetch PF_ROWS ahead (in-bounds).
#pragma unroll 2
    for (int s = 0; s < (KROWS_PER_WAVE - PF_ROWS) / 4; ++s)
        kstep<true>(jp, xp, acc, jsq);
    // Tail: last PF_ROWS rows, no prefetch (keeps addresses in-bounds).
#pragma unroll
    for (int s = 0; s < PF_ROWS / 4; ++s)
        kstep<false>(jp, xp, acc, jsq);

    // Spill accumulators: D-matrix layout is VGPR r -> M = r + 8*half, N = lr
#pragma unroll
    for (int m = 0; m < 4; ++m)
#pragma unroll
        for (int r = 0; r < 8; ++r)
            lds_y[wave][m][(r + half * 8) * 16 + lr] = acc[m][r];
    lds_jsq[tid] = jsq;
    __syncthreads();

    // Reduce across the 4 K-split waves and contract with Xf columns.
    if (tid < BATCH) {
        const int b   = tid;
        const int m   = b >> 4;
        const int row = b & 15;
        float s = 0.0f;
#pragma unroll
        for (int n = 0; n < 16; ++n) {
            const float y = lds_y[0][m][row * 16 + n]
                          + lds_y[1][m][row * 16 + n]
                          + lds_y[2][m][row * 16 + n]
                          + lds_y[3][m][row * 16 + n];
            s += y * X[(size_t)b * DD + col0 + n];
        }
        atomicAdd(&ws_energy[b], s);   // global_atomic_add_f32
    }
    if (tid == 0) {
        float s = 0.0f;
        for (int i = 0; i < 128; ++i) s += lds_jsq[i];
        atomicAdd(ws_regj, s);
    }
}

// ---------------------------------------------------------------------------
// Kernel 2: energy = j_sum + Xf.Hw + Hb; out = sigmoid(energy); reg_h.
// Grid: 65 waves. Blocks 0..63 handle one batch row each; block 64 does regs.
// ---------------------------------------------------------------------------
__global__ __launch_bounds__(32) void potts_finalize_kernel(
    const float* __restrict__ X,
    const float* __restrict__ Hw,
    const float* __restrict__ Hb,
    const float* __restrict__ ws_energy,
    const float* __restrict__ ws_regj,
    float* __restrict__ out)          // [64 sig | 64 energy | reg_j | reg_h]
{
    const int b    = blockIdx.x;
    const int lane = (int)threadIdx.x;

    if (b < BATCH) {
        float dot = 0.0f;
        for (int i = lane; i < DD; i += 32)
            dot += X[(size_t)b * DD + i] * Hw[i];
        for (int off = 16; off > 0; off >>= 1)
            dot += __shfl_xor(dot, off, 32);
        if (lane == 0) {
            const float e = ws_energy[b] + dot + Hb[0];
            out[b]         = 1.0f / (1.0f + expf(-e));
            out[BATCH + b] = e;
        }
    } else {
        float s = 0.0f;
        for (int i = lane; i < DD; i += 32)
            s += Hw[i] * Hw[i];
        for (int off = 16; off > 0; off >>= 1)
            s += __shfl_xor(s, off, 32);
        if (lane == 0) {
            out[2 * BATCH]     = ws_regj[0];  // reg_j
            out[2 * BATCH + 1] = s;           // reg_h
        }
    }
}

// ---------------------------------------------------------------------------
extern "C" void kernel_launch(void* const* d_in, const int* in_sizes, int n_in,
                              void* d_out, int out_size, void* d_ws, size_t ws_size,
                              hipStream_t stream) {
    (void)in_sizes; (void)n_in; (void)out_size; (void)ws_size;
    const float* X  = (const float*)d_in[0];   // [64,128,21] -> [64,2688]
    const float* Jw = (const float*)d_in[1];   // [2688,2688]
    const float* Hw = (const float*)d_in[2];   // [2688]
    const float* Hb = (const float*)d_in[3];   // [1]
    float* out = (float*)d_out;
    float* ws  = (float*)d_ws;                 // ws[0:64]=energy, ws[64]=reg_j

    potts_init_ws<<<1, 128, 0, stream>>>(ws);
    potts_gemm_kernel<<<NTILES, 128, 0, stream>>>(X, Jw, ws, ws + 64);
    potts_finalize_kernel<<<BATCH + 1, 32, 0, stream>>>(X, Hw, Hb, ws, ws + 64, out);
}